// GAT_7687991459902
// MI455X (gfx1250) — compile-verified
//
#include <hip/hip_runtime.h>
#include <math.h>

// GAT 2-layer forward for MI455X (gfx1250, wave32).
// GEMMs use exact-f32 V_WMMA_F32_16X16X4_F32; edge softmax/scatter uses
// native f32 global atomics (memory-bound part, ~3 GB traffic, L2-resident).

typedef __attribute__((ext_vector_type(2))) float v2f;
typedef __attribute__((ext_vector_type(8))) float v8f;

#define NEG_SLOPE 0.2f

// ---- order-preserving float <-> uint mapping for atomic float max ----
__device__ __forceinline__ unsigned f2ord(float f) {
  unsigned u = __float_as_uint(f);
  return (u & 0x80000000u) ? ~u : (u | 0x80000000u);
}
__device__ __forceinline__ float ord2f(unsigned o) {
  return (o & 0x80000000u) ? __uint_as_float(o & 0x7fffffffu)
                           : __uint_as_float(~o);
}

// =====================================================================
// GEMM1: H[N,64] = X[N,128] @ W[128,64]   (one wave -> 16x64 tile)
// =====================================================================
__global__ void gat_gemm1_wmma(const float* __restrict__ X,
                               const float* __restrict__ W,
                               float* __restrict__ H, int ntiles) {
  const int lane = threadIdx.x & 31;
  const int tile = blockIdx.x * (blockDim.x >> 5) + (threadIdx.x >> 5);
  if (tile >= ntiles) return;
  const int row0 = tile * 16;
  const int m = lane & 15;
  const int koff = (lane >> 4) << 1;  // lanes 0-15: K={0,1}; lanes 16-31: K={2,3}

  v8f acc0 = {}, acc1 = {}, acc2 = {}, acc3 = {};
  const float* xr = X + (size_t)(row0 + m) * 128;

  for (int kk = 0; kk < 128; kk += 4) {
    v2f a;
    a.x = xr[kk + koff + 0];
    a.y = xr[kk + koff + 1];
    const float* w0 = W + (size_t)(kk + koff + 0) * 64 + m;
    const float* w1 = W + (size_t)(kk + koff + 1) * 64 + m;
    v2f b0; b0.x = w0[0];  b0.y = w1[0];
    v2f b1; b1.x = w0[16]; b1.y = w1[16];
    v2f b2; b2.x = w0[32]; b2.y = w1[32];
    v2f b3; b3.x = w0[48]; b3.y = w1[48];
    acc0 = __builtin_amdgcn_wmma_f32_16x16x4_f32(false, a, false, b0, (short)0, acc0, false, false);
    acc1 = __builtin_amdgcn_wmma_f32_16x16x4_f32(false, a, false, b1, (short)0, acc1, false, false);
    acc2 = __builtin_amdgcn_wmma_f32_16x16x4_f32(false, a, false, b2, (short)0, acc2, false, false);
    acc3 = __builtin_amdgcn_wmma_f32_16x16x4_f32(false, a, false, b3, (short)0, acc3, false, false);
  }
#pragma unroll
  for (int r = 0; r < 8; ++r) {
    int row = row0 + r + ((lane >> 4) << 3);
    float* hr = H + (size_t)row * 64 + m;
    hr[0]  = acc0[r];
    hr[16] = acc1[r];
    hr[32] = acc2[r];
    hr[48] = acc3[r];
  }
}

// =====================================================================
// GEMM2: H[N,16] = X[N,64] @ W[64,16]   (one wave -> 16x16 tile)
// =====================================================================
__global__ void gat_gemm2_wmma(const float* __restrict__ X,
                               const float* __restrict__ W,
                               float* __restrict__ H, int ntiles) {
  const int lane = threadIdx.x & 31;
  const int tile = blockIdx.x * (blockDim.x >> 5) + (threadIdx.x >> 5);
  if (tile >= ntiles) return;
  const int row0 = tile * 16;
  const int m = lane & 15;
  const int koff = (lane >> 4) << 1;

  v8f acc = {};
  const float* xr = X + (size_t)(row0 + m) * 64;
  for (int kk = 0; kk < 64; kk += 4) {
    v2f a;
    a.x = xr[kk + koff + 0];
    a.y = xr[kk + koff + 1];
    v2f b;
    b.x = W[(size_t)(kk + koff + 0) * 16 + m];
    b.y = W[(size_t)(kk + koff + 1) * 16 + m];
    acc = __builtin_amdgcn_wmma_f32_16x16x4_f32(false, a, false, b, (short)0, acc, false, false);
  }
#pragma unroll
  for (int r = 0; r < 8; ++r) {
    int row = row0 + r + ((lane >> 4) << 3);
    H[(size_t)row * 16 + m] = acc[r];
  }
}

// =====================================================================
// attention projections
// =====================================================================
__global__ void gat_att1(const float* __restrict__ H,
                         const float* __restrict__ att_src,
                         const float* __restrict__ att_dst,
                         float* __restrict__ as_, float* __restrict__ ad_,
                         int total) {  // total = N*8, thread per (node,head)
  int t = blockIdx.x * blockDim.x + threadIdx.x;
  if (t >= total) return;
  int n = t >> 3, h = t & 7;
  const float* hp = H + (size_t)n * 64 + h * 8;
  float s = 0.f, d = 0.f;
#pragma unroll
  for (int c = 0; c < 8; ++c) {
    s += hp[c] * att_src[h * 8 + c];
    d += hp[c] * att_dst[h * 8 + c];
  }
  as_[t] = s;
  ad_[t] = d;
}

__global__ void gat_att2(const float* __restrict__ H,
                         const float* __restrict__ att_src,
                         const float* __restrict__ att_dst,
                         float* __restrict__ as_, float* __restrict__ ad_,
                         int n_nodes) {  // thread per node (H2=1, C2=16)
  int n = blockIdx.x * blockDim.x + threadIdx.x;
  if (n >= n_nodes) return;
  const float* hp = H + (size_t)n * 16;
  float s = 0.f, d = 0.f;
#pragma unroll
  for (int c = 0; c < 16; ++c) {
    s += hp[c] * att_src[c];
    d += hp[c] * att_dst[c];
  }
  as_[n] = s;
  ad_[n] = d;
}

// =====================================================================
// edge src/dst helper: edges [0,E) from edge_index, [E,Etot) self-loops
// =====================================================================
__device__ __forceinline__ void edge_sd(const int* __restrict__ ei,
                                        long long e, long long E,
                                        int& s, int& d) {
  if (e < E) {
    s = ei[e];
    d = ei[E + e];
  } else {
    s = d = (int)(e - E);
  }
}

// ---------------- layer 1 edge passes (H=8, C=8) ---------------------
__global__ void gat_edge1_max(const int* __restrict__ ei,
                              const float* __restrict__ as_,
                              const float* __restrict__ ad_,
                              unsigned* __restrict__ mo,
                              long long E, long long Etot) {
  long long t = (long long)blockIdx.x * blockDim.x + threadIdx.x;
  if (t >= Etot * 8) return;
  long long e = t >> 3;
  int h = (int)(t & 7);
  int s, d;
  edge_sd(ei, e, E, s, d);
  float el = as_[(size_t)s * 8 + h] + ad_[(size_t)d * 8 + h];
  el = el > 0.f ? el : NEG_SLOPE * el;
  atomicMax(&mo[(size_t)d * 8 + h], f2ord(el));
}

__global__ void gat_edge1_sum(const int* __restrict__ ei,
                              const float* __restrict__ as_,
                              const float* __restrict__ ad_,
                              const unsigned* __restrict__ mo,
                              float* __restrict__ z,
                              long long E, long long Etot) {
  long long t = (long long)blockIdx.x * blockDim.x + threadIdx.x;
  if (t >= Etot * 8) return;
  long long e = t >> 3;
  int h = (int)(t & 7);
  int s, d;
  edge_sd(ei, e, E, s, d);
  float el = as_[(size_t)s * 8 + h] + ad_[(size_t)d * 8 + h];
  el = el > 0.f ? el : NEG_SLOPE * el;
  float mm = ord2f(mo[(size_t)d * 8 + h]);
  atomicAdd(&z[(size_t)d * 8 + h], __expf(el - mm));
}

__global__ void gat_edge1_agg(const int* __restrict__ ei,
                              const float* __restrict__ as_,
                              const float* __restrict__ ad_,
                              const unsigned* __restrict__ mo,
                              const float* __restrict__ z,
                              const float* __restrict__ H,
                              float* __restrict__ agg,
                              long long E, long long Etot) {
  long long t = (long long)blockIdx.x * blockDim.x + threadIdx.x;
  if (t >= Etot * 8) return;
  long long e = t >> 3;
  int h = (int)(t & 7);
  int s, d;
  edge_sd(ei, e, E, s, d);
  float el = as_[(size_t)s * 8 + h] + ad_[(size_t)d * 8 + h];
  el = el > 0.f ? el : NEG_SLOPE * el;
  float mm = ord2f(mo[(size_t)d * 8 + h]);
  float alpha = __expf(el - mm) / (z[(size_t)d * 8 + h] + 1e-16f);
  const float* hs = H + (size_t)s * 64 + h * 8;
  float* od = agg + (size_t)d * 64 + h * 8;
#pragma unroll
  for (int c = 0; c < 8; ++c) atomicAdd(&od[c], hs[c] * alpha);
}

// ---------------- layer 2 edge passes (H=1, C=16) --------------------
__global__ void gat_edge2_max(const int* __restrict__ ei,
                              const float* __restrict__ as_,
                              const float* __restrict__ ad_,
                              unsigned* __restrict__ mo,
                              long long E, long long Etot) {
  long long e = (long long)blockIdx.x * blockDim.x + threadIdx.x;
  if (e >= Etot) return;
  int s, d;
  edge_sd(ei, e, E, s, d);
  float el = as_[s] + ad_[d];
  el = el > 0.f ? el : NEG_SLOPE * el;
  atomicMax(&mo[d], f2ord(el));
}

__global__ void gat_edge2_sum(const int* __restrict__ ei,
                              const float* __restrict__ as_,
                              const float* __restrict__ ad_,
                              const unsigned* __restrict__ mo,
                              float* __restrict__ z,
                              long long E, long long Etot) {
  long long e = (long long)blockIdx.x * blockDim.x + threadIdx.x;
  if (e >= Etot) return;
  int s, d;
  edge_sd(ei, e, E, s, d);
  float el = as_[s] + ad_[d];
  el = el > 0.f ? el : NEG_SLOPE * el;
  atomicAdd(&z[d], __expf(el - ord2f(mo[d])));
}

__global__ void gat_edge2_agg(const int* __restrict__ ei,
                              const float* __restrict__ as_,
                              const float* __restrict__ ad_,
                              const unsigned* __restrict__ mo,
                              const float* __restrict__ z,
                              const float* __restrict__ H,
                              float* __restrict__ agg,
                              long long E, long long Etot) {
  long long e = (long long)blockIdx.x * blockDim.x + threadIdx.x;
  if (e >= Etot) return;
  int s, d;
  edge_sd(ei, e, E, s, d);
  float el = as_[s] + ad_[d];
  el = el > 0.f ? el : NEG_SLOPE * el;
  float alpha = __expf(el - ord2f(mo[d])) / (z[d] + 1e-16f);
  const float* hs = H + (size_t)s * 16;
  float* od = agg + (size_t)d * 16;
#pragma unroll
  for (int c = 0; c < 16; ++c) atomicAdd(&od[c], hs[c] * alpha);
}

// =====================================================================
// bias + ELU (in place), and final bias + log_softmax
// =====================================================================
__global__ void gat_bias_elu(float* __restrict__ A, const float* __restrict__ b,
                             int total) {  // total = N*64
  int i = blockIdx.x * blockDim.x + threadIdx.x;
  if (i >= total) return;
  float v = A[i] + b[i & 63];
  A[i] = v > 0.f ? v : expm1f(v);
}

__global__ void gat_final(const float* __restrict__ agg,
                          const float* __restrict__ b2,
                          float* __restrict__ out, int n_nodes) {
  int n = blockIdx.x * blockDim.x + threadIdx.x;
  if (n >= n_nodes) return;
  float v[16];
  float mx = -1e30f;
#pragma unroll
  for (int c = 0; c < 16; ++c) {
    v[c] = agg[(size_t)n * 16 + c] + b2[c];
    mx = fmaxf(mx, v[c]);
  }
  float sum = 0.f;
#pragma unroll
  for (int c = 0; c < 16; ++c) sum += __expf(v[c] - mx);
  float ls = logf(sum) + mx;
#pragma unroll
  for (int c = 0; c < 16; ++c) out[(size_t)n * 16 + c] = v[c] - ls;
}

// =====================================================================
extern "C" void kernel_launch(void* const* d_in, const int* in_sizes, int n_in,
                              void* d_out, int out_size, void* d_ws, size_t ws_size,
                              hipStream_t stream) {
  const float* x        = (const float*)d_in[0];
  const int*   ei       = (const int*)d_in[1];   // [2,E]: row0=src, row1=dst
  const float* W1       = (const float*)d_in[2];
  const float* att_src1 = (const float*)d_in[3];
  const float* att_dst1 = (const float*)d_in[4];
  const float* b1       = (const float*)d_in[5];
  const float* W2       = (const float*)d_in[6];
  const float* att_src2 = (const float*)d_in[7];
  const float* att_dst2 = (const float*)d_in[8];
  const float* b2       = (const float*)d_in[9];

  const int N = in_sizes[0] / 128;
  const long long E = (long long)in_sizes[1] / 2;
  const long long Etot = E + N;

  // ---- carve scratch from d_ws ----
  char* base = (char*)d_ws;
  size_t off = 0;
  auto carve = [&](size_t bytes) -> void* {
    void* p = base + off;
    off = (off + bytes + 255) & ~(size_t)255;
    return p;
  };
  float*    h1   = (float*)carve((size_t)N * 64 * 4);
  float*    as1  = (float*)carve((size_t)N * 8 * 4);
  float*    ad1  = (float*)carve((size_t)N * 8 * 4);
  unsigned* m1   = (unsigned*)carve((size_t)N * 8 * 4);
  float*    z1   = (float*)carve((size_t)N * 8 * 4);
  float*    agg1 = (float*)carve((size_t)N * 64 * 4);
  float*    h2   = (float*)carve((size_t)N * 16 * 4);
  float*    as2  = (float*)carve((size_t)N * 4);
  float*    ad2  = (float*)carve((size_t)N * 4);
  unsigned* m2   = (unsigned*)carve((size_t)N * 4);
  float*    z2   = (float*)carve((size_t)N * 4);
  float*    agg2 = (float*)carve((size_t)N * 16 * 4);
  (void)ws_size;
  (void)n_in;
  (void)out_size;

  const int ntiles = (N + 15) / 16;
  const int wpb = 4;  // waves per block (128 threads)

  // ---- layer 1 ----
  gat_gemm1_wmma<<<(ntiles + wpb - 1) / wpb, wpb * 32, 0, stream>>>(x, W1, h1, ntiles);
  gat_att1<<<(N * 8 + 255) / 256, 256, 0, stream>>>(h1, att_src1, att_dst1, as1, ad1, N * 8);
  hipMemsetAsync(m1, 0, (size_t)N * 8 * 4, stream);
  hipMemsetAsync(z1, 0, (size_t)N * 8 * 4, stream);
  hipMemsetAsync(agg1, 0, (size_t)N * 64 * 4, stream);
  {
    long long T = Etot * 8;
    int g = (int)((T + 255) / 256);
    gat_edge1_max<<<g, 256, 0, stream>>>(ei, as1, ad1, m1, E, Etot);
    gat_edge1_sum<<<g, 256, 0, stream>>>(ei, as1, ad1, m1, z1, E, Etot);
    gat_edge1_agg<<<g, 256, 0, stream>>>(ei, as1, ad1, m1, z1, h1, agg1, E, Etot);
  }
  gat_bias_elu<<<(N * 64 + 255) / 256, 256, 0, stream>>>(agg1, b1, N * 64);

  // ---- layer 2 ----
  gat_gemm2_wmma<<<(ntiles + wpb - 1) / wpb, wpb * 32, 0, stream>>>(agg1, W2, h2, ntiles);
  gat_att2<<<(N + 255) / 256, 256, 0, stream>>>(h2, att_src2, att_dst2, as2, ad2, N);
  hipMemsetAsync(m2, 0, (size_t)N * 4, stream);
  hipMemsetAsync(z2, 0, (size_t)N * 4, stream);
  hipMemsetAsync(agg2, 0, (size_t)N * 16 * 4, stream);
  {
    int g = (int)((Etot + 255) / 256);
    gat_edge2_max<<<g, 256, 0, stream>>>(ei, as2, ad2, m2, E, Etot);
    gat_edge2_sum<<<g, 256, 0, stream>>>(ei, as2, ad2, m2, z2, E, Etot);
    gat_edge2_agg<<<g, 256, 0, stream>>>(ei, as2, ad2, m2, z2, h2, agg2, E, Etot);
  }
  gat_final<<<(N + 255) / 256, 256, 0, stream>>>(agg2, b2, (float*)d_out, N);
}